// AFMAttentionLayer_62319975465590
// MI455X (gfx1250) — compile-verified
//
#include <hip/hip_runtime.h>
#include <stdint.h>

typedef __attribute__((ext_vector_type(2))) float v2f;
typedef __attribute__((ext_vector_type(4))) float v4f;
typedef __attribute__((ext_vector_type(8))) float v8f;
typedef __attribute__((ext_vector_type(4))) unsigned int u32x4;
typedef __attribute__((ext_vector_type(8))) int i32x8;
typedef __attribute__((ext_vector_type(4))) int i32x4;

#define FDIM 64      // F: tokens per batch
#define DDIM 128     // D: feature dim
#define PAD  4       // LDS row padding (16B-aligned rows, conflict-free)
#define LROW (DDIM + PAD)

#if defined(__has_builtin)
#  if __has_builtin(__builtin_amdgcn_tensor_load_to_lds)
#    define HAVE_TDM 1
#  endif
#endif
#ifndef HAVE_TDM
#  define HAVE_TDM 0
#endif

__global__ __launch_bounds__(128)
void afm_attention_kernel(const float* __restrict__ x,
                          const float* __restrict__ w,
                          const float* __restrict__ bb,
                          float* __restrict__ out) {
    __shared__ float xs[FDIM * LROW];   // 64 x (128+4) staged X tile
    __shared__ float ws[DDIM];          // wsum[d] = sum_h w[h][d]
    __shared__ float red[129];          // [0..127] reduction, [128] bsum

    const int tid   = threadIdx.x;
    const int batch = blockIdx.x;
    const float* xb = x + (size_t)batch * FDIM * DDIM;

    // Fold the head dimension: wsum and bsum.
    const int d = tid & (DDIM - 1);
    ws[d] = w[d] + w[DDIM + d] + w[2 * DDIM + d] + w[3 * DDIM + d];
    if (tid == 0) red[128] = bb[0] + bb[1] + bb[2] + bb[3];

#if HAVE_TDM
    // Tensor Data Mover: DMA the 64x128 f32 tile into LDS, hardware-inserting
    // 16B of padding after every 128-DWORD row (matches LROW = 128+4 layout).
    if (tid < 32) {   // one TDM op per block, issued by wave 0 (EXEC ignored)
        unsigned lds_off = (unsigned)(unsigned long long)(uintptr_t)(void*)xs;
        unsigned long long ga = (unsigned long long)(uintptr_t)(const void*)xb;
        // D# group 0: count=1 | lds_addr | global_addr[56:0] | type=2
        u32x4 g0 = { 1u,
                     lds_off,
                     (unsigned)(ga & 0xFFFFFFFFull),
                     (unsigned)((ga >> 32) & 0x01FFFFFFull) | 0x80000000u };
        // D# group 1: data_size=4B(2), pad_enable, pad_interval=6 (128 DW),
        // pad_amount=3 (4 DW); tensor 128x64, tile 128x64, dim0_stride=128.
        i32x8 g1 = { (int)((2u << 16) | (1u << 20) | (6u << 22) | (3u << 25)),
                     (int)(128u << 16),            // tensor_dim0 = 128 (bits 79:48)
                     (int)(64u << 16),             // tensor_dim1 = 64  (bits 111:80)
                     (int)(128u << 16),            // tile_dim0   = 128 (bits 127:112)
                     64,                           // tile_dim1   = 64  (bits 143:128)
                     128,                          // tensor_dim0_stride = 128
                     (int)(0x2000u << 16),         // tensor_dim1_stride = 8192
                     0 };
        i32x4 gz = { 0, 0, 0, 0 };
#if __clang_major__ >= 23
        i32x8 gz8 = { 0, 0, 0, 0, 0, 0, 0, 0 };
        __builtin_amdgcn_tensor_load_to_lds(g0, g1, gz, gz, gz8, 0);
#else
        __builtin_amdgcn_tensor_load_to_lds(g0, g1, gz, gz, 0);
#endif
        __builtin_amdgcn_s_wait_tensorcnt(0);
    }
#else
    // Fallback: manual staging, 16 float4 per thread.
    #pragma unroll
    for (int i = 0; i < 16; ++i) {
        int idx = tid + i * 128;
        int row = idx >> 5;
        int c4  = idx & 31;
        v4f v = *(const v4f*)(xb + row * DDIM + c4 * 4);
        *(v4f*)(xs + row * LROW + c4 * 4) = v;
    }
#endif
    __syncthreads();

    const int wave  = tid >> 5;         // 0..3 (wave32)
    const int lane  = tid & 31;
    const int l16   = lane & 15;
    const int khalf = (lane >> 4) * 2;  // 0 or 2: K-pair this lane holds

    const float bsum = red[128];
    const float* arow = xs + (wave * 16 + l16) * LROW;

    v8f acc[4] = {v8f{0}, v8f{0}, v8f{0}, v8f{0}};

    // S_tile = (A * diag(wsum)) * B, K=128 in steps of 4 via V_WMMA_F32_16X16X4_F32.
    #pragma unroll 4
    for (int k0 = 0; k0 < DDIM; k0 += 4) {
        const int k = k0 + khalf;
        v2f a  = *(const v2f*)(arow + k);
        v2f wv = *(const v2f*)(ws + k);
        a.x *= wv.x;
        a.y *= wv.y;
        #pragma unroll
        for (int c = 0; c < 4; ++c) {
            v2f bv = *(const v2f*)(xs + (c * 16 + l16) * LROW + k);  // (X^T)[k][n] = X[n][k]
            acc[c] = __builtin_amdgcn_wmma_f32_16x16x4_f32(
                         /*neg_a=*/false, a, /*neg_b=*/false, bv,
                         /*c_mod=*/(short)0, acc[c],
                         /*reuse_a=*/false, /*reuse_b=*/false);
        }
    }

    // Denominator: sum over all 64x64 entries of (acc + bsum).
    float p = 0.f;
    #pragma unroll
    for (int c = 0; c < 4; ++c)
        #pragma unroll
        for (int e = 0; e < 8; ++e)
            p += acc[c][e] + bsum;
    red[tid] = p;
    __syncthreads();
    #pragma unroll
    for (int off = 64; off >= 1; off >>= 1) {
        if (tid < off) red[tid] += red[tid + off];
        __syncthreads();
    }
    const float inv = 1.0f / red[0];

    // Store: C/D layout — VGPR e holds row (e + 8*(lane>=16)), col = lane&15.
    float* ob = out + (size_t)batch * FDIM * FDIM;
    const int mbase = wave * 16 + (lane >> 4) * 8;
    #pragma unroll
    for (int c = 0; c < 4; ++c) {
        #pragma unroll
        for (int e = 0; e < 8; ++e) {
            ob[(mbase + e) * FDIM + c * 16 + l16] = (acc[c][e] + bsum) * inv;
        }
    }
}

extern "C" void kernel_launch(void* const* d_in, const int* in_sizes, int n_in,
                              void* d_out, int out_size, void* d_ws, size_t ws_size,
                              hipStream_t stream) {
    const float* x  = (const float*)d_in[0];   // [B, 64, 128] f32
    const float* w  = (const float*)d_in[1];   // [4, 128] f32
    const float* bb = (const float*)d_in[2];   // [4] f32
    float* out      = (float*)d_out;           // [B, 64, 64] f32

    const int B = in_sizes[0] / (FDIM * DDIM); // 4096
    afm_attention_kernel<<<B, 128, 0, stream>>>(x, w, bb, out);
}